// BatchelorAdj_43301860278792
// MI455X (gfx1250) — compile-verified
//
#include <hip/hip_runtime.h>
#include <math.h>

// ---------------------------------------------------------------------------
// MRI motion-compensated adjoint on MI455X (gfx1250, wave32, WMMA).
// ifft2c(K) = A * K * A with centered inverse-DFT matrix A (fp32 complex),
// implemented as fp32 WMMA (V_WMMA_F32_16X16X4_F32) complex GEMMs.
// Each wave computes a 32x32 complex tile (2x2 WMMA subtiles): 16 WMMAs per
// K=4 step vs 8 contiguous global_load_b64 fragment loads. Operand addressing
// is pure pointer-increment so the hot loop carries no 64-bit index math.
// ---------------------------------------------------------------------------

typedef float v2f __attribute__((ext_vector_type(2)));
typedef float v8f __attribute__((ext_vector_type(8)));

#define NX 320
#define N2 (NX * NX)
#define NC 16
#define NT 25

// ---------------------------------------------------------------------------
__global__ void zero_f32(float* __restrict__ p, int n) {
  int i = blockIdx.x * blockDim.x + threadIdx.x;
  if (i < n) p[i] = 0.0f;
}

// ---------------------------------------------------------------------------
// Centered inverse-DFT matrix: A[m,n] = exp(+2*pi*i*(m-160)*(n-160)/320)/sqrt(320)
// A is symmetric, so its column-major view is the same array.
__global__ void build_dft(float* __restrict__ Are, float* __restrict__ Aim) {
  int idx = blockIdx.x * blockDim.x + threadIdx.x;
  if (idx >= N2) return;
  int m = idx / NX, n = idx % NX;
  int r = ((m - 160) * (n - 160)) % NX;
  if (r < 0) r += NX;
  float th = 6.283185307179586f * (float)r / (float)NX;
  float s, c;
  __sincosf(th, &s, &c);
  const float inv = 0.05590169943749474f;  // 1/sqrt(320)
  Are[idx] = c * inv;
  Aim[idx] = s * inv;
}

// ---------------------------------------------------------------------------
// Gather masked k-space for frame t into per-coil COLUMN-MAJOR planes:
// KmT[c][y][x] = kspace[x,y,c] * mask[x,y,c,t]   (B operand of GEMM1, so the
// K dimension (x) is contiguous -> b64 fragment loads).
__global__ void build_km(const float* __restrict__ kr, const float* __restrict__ ki,
                         const float* __restrict__ mask, int t,
                         float* __restrict__ KreT, float* __restrict__ KimT) {
  int idx = blockIdx.x * blockDim.x + threadIdx.x;  // idx = p*16 + c
  if (idx >= N2 * NC) return;
  int c = idx & (NC - 1);
  int p = idx >> 4;  // x*320 + y
  int x = p / NX;
  int y = p - x * NX;
  float mv = mask[idx * NT + t];
  int o = c * N2 + y * NX + x;  // transposed
  KreT[o] = kr[idx] * mv;
  KimT[o] = ki[idx] * mv;
}

// ---------------------------------------------------------------------------
// One wave computes a 32x32 complex tile of C = Ac * Bc.
// Ac: row-major re/im planes (K contiguous). BcT: column-major re/im planes
// (K contiguous). 2x2 subtiles of 16x16; 16 fp32 WMMAs per K=4 step:
//   c_re += a_re*b_re + (-a_im)*b_im ;  c_im += a_re*b_im + a_im*b_re
__device__ __forceinline__ void cgemm32(
    const float* __restrict__ Are, const float* __restrict__ Aim,
    const float* __restrict__ BreT, const float* __restrict__ BimT,
    int m0, int n0, v8f (&cre)[2][2], v8f (&cim)[2][2]) {
  const int lane = threadIdx.x & 31;
  const int half = lane >> 4;  // 0: lanes 0-15 (K 0,1), 1: lanes 16-31 (K 2,3)
  const int l16 = lane & 15;
  const int kofs = 2 * half;   // even -> 8B-aligned v2f loads

  // Eight per-lane operand stream pointers, advanced by pointer-increment only.
  const float* pare0 = Are + (m0 + l16) * NX + kofs;
  const float* pare1 = Are + (m0 + 16 + l16) * NX + kofs;
  const float* paim0 = Aim + (m0 + l16) * NX + kofs;
  const float* paim1 = Aim + (m0 + 16 + l16) * NX + kofs;
  const float* pbre0 = BreT + (n0 + l16) * NX + kofs;
  const float* pbre1 = BreT + (n0 + 16 + l16) * NX + kofs;
  const float* pbim0 = BimT + (n0 + l16) * NX + kofs;
  const float* pbim1 = BimT + (n0 + 16 + l16) * NX + kofs;

#pragma unroll 2
  for (int k = 0; k < NX; k += 4) {
    if ((k & 63) == 0) {  // uniform: prefetch ~80 floats (320B) ahead
      __builtin_prefetch(pare0 + 80, 0, 0);
      __builtin_prefetch(pare1 + 80, 0, 0);
      __builtin_prefetch(paim0 + 80, 0, 0);
      __builtin_prefetch(paim1 + 80, 0, 0);
      __builtin_prefetch(pbre0 + 80, 0, 0);
      __builtin_prefetch(pbre1 + 80, 0, 0);
      __builtin_prefetch(pbim0 + 80, 0, 0);
      __builtin_prefetch(pbim1 + 80, 0, 0);
    }

    v2f are0 = *(const v2f*)pare0;
    v2f are1 = *(const v2f*)pare1;
    v2f aim0 = *(const v2f*)paim0;
    v2f aim1 = *(const v2f*)paim1;
    v2f bre0 = *(const v2f*)pbre0;
    v2f bre1 = *(const v2f*)pbre1;
    v2f bim0 = *(const v2f*)pbim0;
    v2f bim1 = *(const v2f*)pbim1;
    v2f aimn0 = -aim0;
    v2f aimn1 = -aim1;

#define CWMMA(MI, NI, ARE, AIM, AIMN, BRE, BIM)                                \
  cre[MI][NI] = __builtin_amdgcn_wmma_f32_16x16x4_f32(                         \
      false, ARE, false, BRE, (short)0, cre[MI][NI], false, false);            \
  cre[MI][NI] = __builtin_amdgcn_wmma_f32_16x16x4_f32(                         \
      false, AIMN, false, BIM, (short)0, cre[MI][NI], false, false);           \
  cim[MI][NI] = __builtin_amdgcn_wmma_f32_16x16x4_f32(                         \
      false, ARE, false, BIM, (short)0, cim[MI][NI], false, false);            \
  cim[MI][NI] = __builtin_amdgcn_wmma_f32_16x16x4_f32(                         \
      false, AIM, false, BRE, (short)0, cim[MI][NI], false, false);

    CWMMA(0, 0, are0, aim0, aimn0, bre0, bim0)
    CWMMA(0, 1, are0, aim0, aimn0, bre1, bim1)
    CWMMA(1, 0, are1, aim1, aimn1, bre0, bim0)
    CWMMA(1, 1, are1, aim1, aimn1, bre1, bim1)
#undef CWMMA

    pare0 += 4;
    pare1 += 4;
    paim0 += 4;
    paim1 += 4;
    pbre0 += 4;
    pbre1 += 4;
    pbim0 += 4;
    pbim1 += 4;
  }
}

// ---------------------------------------------------------------------------
// GEMM1 (batched over coil): M1[c] = A * Km[c]   (complex, 320x320x320)
// M1 stored row-major (it is the A operand of GEMM2).
__global__ void gemm1(const float* __restrict__ Are, const float* __restrict__ Aim,
                      const float* __restrict__ KreT, const float* __restrict__ KimT,
                      float* __restrict__ M1re, float* __restrict__ M1im) {
  const int c = blockIdx.z;
  const int m0 = blockIdx.y * 32;
  const int n0 = blockIdx.x * 32;
  v8f cre[2][2] = {};
  v8f cim[2][2] = {};
  cgemm32(Are, Aim, KreT + c * N2, KimT + c * N2, m0, n0, cre, cim);

  const int lane = threadIdx.x & 31;
  const int half = lane >> 4;
  const int l16 = lane & 15;
#pragma unroll
  for (int mi = 0; mi < 2; ++mi)
#pragma unroll
    for (int ni = 0; ni < 2; ++ni)
#pragma unroll
      for (int r = 0; r < 8; ++r) {
        int m = m0 + mi * 16 + r + 8 * half;
        int n = n0 + ni * 16 + l16;
        M1re[c * N2 + m * NX + n] = cre[mi][ni][r];
        M1im[c * N2 + m * NX + n] = cim[mi][ni][r];
      }
}

// ---------------------------------------------------------------------------
// GEMM2 + coil combine: imaux = sum_c conj(smap_c) .* (M1[c] * A)
// B operand is the DFT matrix A: symmetric, so Are/Aim serve as their own
// column-major planes.
__global__ void gemm2(const float* __restrict__ M1re, const float* __restrict__ M1im,
                      const float* __restrict__ Are, const float* __restrict__ Aim,
                      const float* __restrict__ sr, const float* __restrict__ si,
                      float* __restrict__ outre, float* __restrict__ outim) {
  const int m0 = blockIdx.y * 32;
  const int n0 = blockIdx.x * 32;
  const int lane = threadIdx.x & 31;
  const int half = lane >> 4;
  const int l16 = lane & 15;

  v8f accre[2][2] = {};
  v8f accim[2][2] = {};

  for (int c = 0; c < NC; ++c) {
    v8f cre[2][2] = {};
    v8f cim[2][2] = {};
    cgemm32(M1re + c * N2, M1im + c * N2, Are, Aim, m0, n0, cre, cim);
#pragma unroll
    for (int mi = 0; mi < 2; ++mi)
#pragma unroll
      for (int ni = 0; ni < 2; ++ni)
#pragma unroll
        for (int r = 0; r < 8; ++r) {
          int m = m0 + mi * 16 + r + 8 * half;
          int n = n0 + ni * 16 + l16;
          float s_re = sr[(m * NX + n) * NC + c];
          float s_im = si[(m * NX + n) * NC + c];
          // conj(s)*z = (s_re*z_re + s_im*z_im) + i*(s_re*z_im - s_im*z_re)
          accre[mi][ni][r] =
              fmaf(s_re, cre[mi][ni][r], fmaf(s_im, cim[mi][ni][r], accre[mi][ni][r]));
          accim[mi][ni][r] =
              fmaf(s_re, cim[mi][ni][r], fmaf(-s_im, cre[mi][ni][r], accim[mi][ni][r]));
        }
  }

#pragma unroll
  for (int mi = 0; mi < 2; ++mi)
#pragma unroll
    for (int ni = 0; ni < 2; ++ni)
#pragma unroll
      for (int r = 0; r < 8; ++r) {
        int m = m0 + mi * 16 + r + 8 * half;
        int n = n0 + ni * 16 + l16;
        outre[m * NX + n] = accre[mi][ni][r];
        outim[m * NX + n] = accim[mi][ni][r];
      }
}

// ---------------------------------------------------------------------------
// Adjoint bilinear warp for frame t: scatter-add each pixel's value to the 4
// bilinear neighbors of its displaced location; accumulate into d_out [Nx,Ny,2].
__global__ void warp_adjoint(const float* __restrict__ imre, const float* __restrict__ imim,
                             const float* __restrict__ flow, int t,
                             float* __restrict__ out) {
  int idx = blockIdx.x * blockDim.x + threadIdx.x;
  if (idx >= N2) return;
  int x = idx / NX, y = idx % NX;
  float u = flow[(idx * 2 + 0) * NT + t];
  float v = flow[(idx * 2 + 1) * NT + t];
  float px = fminf(fmaxf((float)x + u, 0.0f), (float)(NX - 1));
  float py = fminf(fmaxf((float)y + v, 0.0f), (float)(NX - 1));
  int x0 = (int)floorf(px);
  int y0 = (int)floorf(py);
  int x1 = min(x0 + 1, NX - 1);
  int y1 = min(y0 + 1, NX - 1);
  float wx = px - (float)x0;
  float wy = py - (float)y0;
  float zr = imre[idx], zi = imim[idx];

  float w00 = (1.0f - wx) * (1.0f - wy);
  float w01 = (1.0f - wx) * wy;
  float w10 = wx * (1.0f - wy);
  float w11 = wx * wy;

  atomicAdd(&out[(x0 * NX + y0) * 2 + 0], w00 * zr);
  atomicAdd(&out[(x0 * NX + y0) * 2 + 1], w00 * zi);
  atomicAdd(&out[(x0 * NX + y1) * 2 + 0], w01 * zr);
  atomicAdd(&out[(x0 * NX + y1) * 2 + 1], w01 * zi);
  atomicAdd(&out[(x1 * NX + y0) * 2 + 0], w10 * zr);
  atomicAdd(&out[(x1 * NX + y0) * 2 + 1], w10 * zi);
  atomicAdd(&out[(x1 * NX + y1) * 2 + 0], w11 * zr);
  atomicAdd(&out[(x1 * NX + y1) * 2 + 1], w11 * zi);
}

// ---------------------------------------------------------------------------
extern "C" void kernel_launch(void* const* d_in, const int* in_sizes, int n_in,
                              void* d_out, int out_size, void* d_ws, size_t ws_size,
                              hipStream_t stream) {
  const float* kr = (const float*)d_in[0];    // [320,320,16]
  const float* ki = (const float*)d_in[1];    // [320,320,16]
  const float* mask = (const float*)d_in[2];  // [320,320,16,25]
  const float* sr = (const float*)d_in[3];    // [320,320,16]
  const float* si = (const float*)d_in[4];    // [320,320,16]
  const float* flow = (const float*)d_in[5];  // [320,320,2,25]
  float* out = (float*)d_out;                 // [320,320,2]
  float* ws = (float*)d_ws;

  // Workspace layout (floats): 68 * N2 = ~27.9 MB
  float* Are = ws;               // N2
  float* Aim = Are + N2;         // N2
  float* KreT = Aim + N2;        // 16*N2 (column-major per coil)
  float* KimT = KreT + NC * N2;  // 16*N2
  float* M1re = KimT + NC * N2;  // 16*N2 (row-major per coil)
  float* M1im = M1re + NC * N2;  // 16*N2
  float* imre = M1im + NC * N2;  // N2
  float* imim = imre + N2;       // N2

  zero_f32<<<(2 * N2 + 255) / 256, 256, 0, stream>>>(out, 2 * N2);
  build_dft<<<(N2 + 255) / 256, 256, 0, stream>>>(Are, Aim);

  dim3 g1(NX / 32, NX / 32, NC);
  dim3 g2(NX / 32, NX / 32, 1);

  for (int t = 0; t < NT; ++t) {
    build_km<<<(N2 * NC + 255) / 256, 256, 0, stream>>>(kr, ki, mask, t, KreT, KimT);
    gemm1<<<g1, 32, 0, stream>>>(Are, Aim, KreT, KimT, M1re, M1im);
    gemm2<<<g2, 32, 0, stream>>>(M1re, M1im, Are, Aim, sr, si, imre, imim);
    warp_adjoint<<<(N2 + 255) / 256, 256, 0, stream>>>(imre, imim, flow, t, out);
  }
}